// GCNLayer_89799176225551
// MI455X (gfx1250) — compile-verified
//
#include <hip/hip_runtime.h>

// GCN layer: out = diag(dinv) * A * diag(dinv) * (values @ W + b)
// N = 8192 nodes, D = OUT = 128 features. All f32.

typedef __attribute__((ext_vector_type(2))) float v2f;
typedef __attribute__((ext_vector_type(8))) float v8f;

#define NN   8192
#define DD   128
#define OO   128
#define MTILE 32
#define KTILE 32

// ---------------------------------------------------------------------------
// Kernel 1: dinv[row] = 1 / (sqrt(sum_j A[row][j]) + 1e-8)
// One block per row; 256 threads stream 8192 floats as float4, LDS reduce.
// ---------------------------------------------------------------------------
__global__ __launch_bounds__(256) void rowsum_rsqrt_kernel(
    const float* __restrict__ A, float* __restrict__ dinv) {
  __shared__ float red[256];
  const int row = blockIdx.x;
  const float4* Arow = (const float4*)(A + (size_t)row * NN);
  float s = 0.0f;
  for (int i = threadIdx.x; i < NN / 4; i += 256) {
    float4 v = Arow[i];
    s += (v.x + v.y) + (v.z + v.w);
  }
  red[threadIdx.x] = s;
  __syncthreads();
  for (int off = 128; off > 0; off >>= 1) {
    if (threadIdx.x < off) red[threadIdx.x] += red[threadIdx.x + off];
    __syncthreads();
  }
  if (threadIdx.x == 0) dinv[row] = 1.0f / (sqrtf(red[0]) + 1e-8f);
}

// ---------------------------------------------------------------------------
// Kernel 2: fcs[j][o] = dinv[j] * (sum_d values[j][d]*W[d][o] + b[o])
// 0.27 GFLOP; coalesced over o within a wave, values row broadcast from cache.
// ---------------------------------------------------------------------------
__global__ __launch_bounds__(256) void fc_scale_kernel(
    const float* __restrict__ values, const float* __restrict__ W,
    const float* __restrict__ b, const float* __restrict__ dinv,
    float* __restrict__ fcs) {
  const int idx = blockIdx.x * 256 + threadIdx.x;
  const int j = idx >> 7;    // node row
  const int o = idx & 127;   // output feature
  const float* vr = values + (size_t)j * DD;
  float acc = b[o];
#pragma unroll 8
  for (int d = 0; d < DD; ++d) acc += vr[d] * W[d * OO + o];
  fcs[idx] = acc * dinv[j];
}

// ---------------------------------------------------------------------------
// Kernel 3: out[i][:] = dinv[i] * (A[i][:] @ fcs)   (the 17.2 GFLOP GEMM)
// Block = 32-row panel x 128 cols, 8 waves. Wave: 16 rows x 32 cols
// (two v8f accumulators) via V_WMMA_F32_16X16X4_F32. K staged through LDS.
// ---------------------------------------------------------------------------
__global__ __launch_bounds__(256) void gcn_spmm_kernel(
    const float* __restrict__ A,    // [NN][NN]
    const float* __restrict__ Bm,   // fcs [NN][OO]
    const float* __restrict__ dinv, // [NN]
    float* __restrict__ Cm) {       // out [NN][OO]
  __shared__ float As[MTILE][KTILE + 4];  // 32 x 36 (pad -> conflict-free frags)
  __shared__ float Bs[KTILE][OO + 4];     // 32 x 132

  const int tid  = threadIdx.x;
  const int lane = tid & 31;
  const int wave = tid >> 5;
  const int half = lane >> 4;   // 0 / 1
  const int l    = lane & 15;
  const int wrow = (wave >> 2) * 16;  // 0 or 16 : wave's row group in panel
  const int wcol = (wave & 3) * 32;   // 0,32,64,96 : wave's col group
  const int panel = blockIdx.x * MTILE;

  v8f acc0 = {};
  v8f acc1 = {};

  // Global->LDS load mapping (float4 per thread for A; 4x float4 for B)
  const int a_row = tid >> 3;                 // 0..31
  const int a_kq  = (tid & 7) * 4;            // col within K tile
  const size_t a_base = (size_t)(panel + a_row) * NN + a_kq;

  for (int kb = 0; kb < NN; kb += KTILE) {
    // ---- stage A tile: 32 rows x 32 K values (one float4 per thread)
    {
      float4 av = *(const float4*)(A + a_base + kb);
      As[a_row][a_kq + 0] = av.x;
      As[a_row][a_kq + 1] = av.y;
      As[a_row][a_kq + 2] = av.z;
      As[a_row][a_kq + 3] = av.w;
      if (kb + KTILE < NN)  // prefetch next A tile line (global_prefetch_b8)
        __builtin_prefetch(A + a_base + kb + KTILE, 0, 3);
    }
    // ---- stage B tile: 32 rows x 128 cols (4 float4 per thread)
#pragma unroll
    for (int i = 0; i < 4; ++i) {
      const int f  = tid + i * 256;
      const int br = f >> 5;         // 0..31
      const int bc = (f & 31) * 4;   // 0..124
      float4 bv = *(const float4*)(Bm + (size_t)(kb + br) * OO + bc);
      Bs[br][bc + 0] = bv.x;
      Bs[br][bc + 1] = bv.y;
      Bs[br][bc + 2] = bv.z;
      Bs[br][bc + 3] = bv.w;
    }
    __syncthreads();

    // ---- 8 K-steps of 4: two WMMAs per step (shared A fragment)
#pragma unroll
    for (int kk = 0; kk < KTILE / 4; ++kk) {
      const int kf = kk * 4 + 2 * half;
      v2f a;  // A frag: lane l holds M=l, K = {kf, kf+1}   (vgpr v -> K=v+2*half)
      a.x = As[wrow + l][kf + 0];
      a.y = As[wrow + l][kf + 1];
      v2f b0, b1;  // B frag: lane l holds N=l, K = {kf, kf+1}
      b0.x = Bs[kf + 0][wcol + l];
      b0.y = Bs[kf + 1][wcol + l];
      b1.x = Bs[kf + 0][wcol + 16 + l];
      b1.y = Bs[kf + 1][wcol + 16 + l];
      acc0 = __builtin_amdgcn_wmma_f32_16x16x4_f32(
          false, a, false, b0, (short)0, acc0, false, false);
      acc1 = __builtin_amdgcn_wmma_f32_16x16x4_f32(
          false, a, false, b1, (short)0, acc1, false, false);
    }
    __syncthreads();
  }

  // ---- epilogue: C layout M = vgpr + 8*half, N = l; scale by dinv[row]
#pragma unroll
  for (int v = 0; v < 8; ++v) {
    const int row = panel + wrow + v + 8 * half;
    const float s = dinv[row];
    Cm[(size_t)row * OO + wcol + l]      = acc0[v] * s;
    Cm[(size_t)row * OO + wcol + 16 + l] = acc1[v] * s;
  }
}

// ---------------------------------------------------------------------------
extern "C" void kernel_launch(void* const* d_in, const int* in_sizes, int n_in,
                              void* d_out, int out_size, void* d_ws, size_t ws_size,
                              hipStream_t stream) {
  const float* values    = (const float*)d_in[0];  // [8192,128]
  const float* adjacency = (const float*)d_in[1];  // [8192,8192]
  const float* W         = (const float*)d_in[2];  // [128,128]
  const float* b         = (const float*)d_in[3];  // [128]
  float* out = (float*)d_out;                      // [8192,128]

  float* dinv = (float*)d_ws;                               // 8192 f32 (32 KB)
  float* fcs  = (float*)((char*)d_ws + NN * sizeof(float)); // 8192x128 f32 (4 MB)

  rowsum_rsqrt_kernel<<<NN, 256, 0, stream>>>(adjacency, dinv);
  fc_scale_kernel<<<(NN * OO) / 256, 256, 0, stream>>>(values, W, b, dinv, fcs);
  gcn_spmm_kernel<<<NN / MTILE, 256, 0, stream>>>(adjacency, fcs, dinv, out);
}